// PershomBase_6828998001465
// MI455X (gfx1250) — compile-verified
//
#include <hip/hip_runtime.h>
#include <math.h>

typedef float v2f __attribute__((ext_vector_type(2)));
typedef float v8f __attribute__((ext_vector_type(8)));
typedef unsigned long long u64;

#define NUM_GRAPHS 64
#define NODES 256
#define EDGES 1024
#define D_FEAT 512
#define H_FIL 1024
#define APAD 516     // 512 + 4 pad -> conflict-free strided column reads (516 % 64 == 4)
#define ROWS 64      // rows of x per block (4 M-tiles, register-blocked)
#define WAVES 8      // 256 threads

__device__ __forceinline__ unsigned ordKey(float x) {
  unsigned b = __float_as_uint(x);
  return (b & 0x80000000u) ? ~b : (b | 0x80000000u);
}
__device__ __forceinline__ float ordInv(unsigned k) {
  unsigned b = (k & 0x80000000u) ? (k ^ 0x80000000u) : ~k;
  return __uint_as_float(b);
}

// ---------------------------------------------------------------------------
// Kernel A: f = sigmoid( relu(x @ W1 + b1) @ W2 + b2 ), fused.
// 256 blocks x 256 threads. Each block owns 64 rows of x (staged in 132KB of
// dynamic LDS); each wave owns 8 of the 64 N-tiles. Each B fragment fetched
// from L2 feeds 4 fp32 WMMAs (one per 16-row M-tile): 4x less W1 traffic.
// ---------------------------------------------------------------------------
__global__ __launch_bounds__(256) void fil_gemm_sigmoid(
    const float* __restrict__ x, const float* __restrict__ w1,
    const float* __restrict__ b1, const float* __restrict__ w2,
    const float* __restrict__ b2, float* __restrict__ f_out) {
  extern __shared__ float smem[];
  float* As   = smem;                 // [ROWS][APAD]
  float* fsum = smem + ROWS * APAD;   // [WAVES][ROWS]

  const int t = threadIdx.x;
  const int row0 = blockIdx.x * ROWS;

  // Stage A tile (64 x 512 fp32) into LDS, float4-coalesced.
  for (int idx = t; idx < ROWS * 128; idx += 256) {
    const int r = idx >> 7, c4 = idx & 127;
    const float4 v = ((const float4*)(x + (size_t)(row0 + r) * D_FEAT))[c4];
    float* d = &As[r * APAD + c4 * 4];
    d[0] = v.x; d[1] = v.y; d[2] = v.z; d[3] = v.w;
  }
  __syncthreads();

  const int wave = t >> 5, lane = t & 31;
  const int m = lane & 15;        // row within M-tile / column within B tile
  const int half = lane >> 4;     // 0: K 0..1, 1: K 2..3
  const int koff = half << 1;

  float facc[4][8];
#pragma unroll
  for (int mt = 0; mt < 4; mt++)
#pragma unroll
    for (int r = 0; r < 8; r++) facc[mt][r] = 0.f;

  for (int nt = wave; nt < H_FIL / 16; nt += WAVES) {
    const int col = nt * 16 + m;
    const float* wp = w1 + col;  // stride H_FIL per k
    v8f c[4];
    const v8f zv{};
#pragma unroll
    for (int mt = 0; mt < 4; mt++) c[mt] = zv;

#pragma unroll 4
    for (int kt = 0; kt < D_FEAT / 4; kt++) {
      const int kk = kt * 4 + koff;
      v2f b;
      b.x = wp[(size_t)kk * H_FIL];
      b.y = wp[(size_t)(kk + 1) * H_FIL];
#pragma unroll
      for (int mt = 0; mt < 4; mt++) {
        v2f a;
        a.x = As[(mt * 16 + m) * APAD + kk];
        a.y = As[(mt * 16 + m) * APAD + kk + 1];
        c[mt] = __builtin_amdgcn_wmma_f32_16x16x4_f32(
            /*neg_a=*/false, a, /*neg_b=*/false, b,
            /*c_mod=*/(short)0, c[mt], /*reuse_a=*/false, /*reuse_b=*/false);
      }
    }
    // bias + relu + dot with w_fil2 column weights, accumulated per output row
    const float bias = b1[col], wc = w2[col];
#pragma unroll
    for (int mt = 0; mt < 4; mt++)
#pragma unroll
      for (int r = 0; r < 8; r++) facc[mt][r] += fmaxf(c[mt][r] + bias, 0.f) * wc;
  }

  // Reduce over the 16 N-lanes of each half-wave (xor masks < 16 stay in-half).
#pragma unroll
  for (int mt = 0; mt < 4; mt++)
#pragma unroll
    for (int r = 0; r < 8; r++) {
      float v = facc[mt][r];
#pragma unroll
      for (int s = 1; s < 16; s <<= 1) v += __shfl_xor(v, s, 32);
      if (m == 0) fsum[wave * ROWS + mt * 16 + r + half * 8] = v;
    }
  __syncthreads();
  if (t < ROWS) {  // deterministic fixed-order cross-wave sum
    float pre = b2[0];
#pragma unroll
    for (int w = 0; w < WAVES; w++) pre += fsum[w * ROWS + t];
    f_out[row0 + t] = 1.0f / (1.0f + __expf(-pre));
  }
}

// ---------------------------------------------------------------------------
// Kernel B: per-graph 0-dim persistence (sub + super level) + phi MLPs +
// segment-sum pooling + linear head. One workgroup (256 threads) per graph.
// ---------------------------------------------------------------------------
__global__ __launch_bounds__(256) void persistence_cls(
    const float* __restrict__ f_all, const int* __restrict__ edges,
    const float* __restrict__ w_phi0, const float* __restrict__ b_phi0,
    const float* __restrict__ w_phi1, const float* __restrict__ b_phi1,
    const float* __restrict__ w_head, const float* __restrict__ b_head,
    float* __restrict__ out) {
  __shared__ u64 keys[EDGES];
  __shared__ float lf[NODES], lg[NODES], death[NODES];
  __shared__ float dpair[2][NODES];      // deaths for pass 0 (f) and pass 1 (-f)
  __shared__ int parent[NODES], paired[NODES];
  __shared__ unsigned cmax[NODES];
  __shared__ float pbuf[2][NODES];

  const int g = blockIdx.x, t = threadIdx.x;
  lf[t] = f_all[g * NODES + t];
  __syncthreads();

  for (int pass = 0; pass < 2; pass++) {
    lg[t] = pass ? -lf[t] : lf[t];
    parent[t] = t; death[t] = 0.f; paired[t] = 0; cmax[t] = 0u;
    __syncthreads();

    // Build sort keys: (ordered(w) << 32) | idx<<16 | u<<8 | v.
    // Embedded idx reproduces jnp.argsort stability; keys are unique.
    for (int i = t; i < EDGES; i += 256) {
      const int2 e = ((const int2*)edges)[g * EDGES + i];
      const int u = e.x & (NODES - 1), v = e.y & (NODES - 1);
      const float w = fmaxf(lg[u], lg[v]);
      keys[i] = ((u64)ordKey(w) << 32) | ((u64)i << 16) | ((u64)u << 8) | (u64)v;
    }

    // Bitonic sort, ascending, 1024 keys with 256 threads.
    for (unsigned size = 2; size <= EDGES; size <<= 1)
      for (unsigned stride = size >> 1; stride > 0; stride >>= 1) {
        __syncthreads();
        for (unsigned i = t; i < EDGES / 2; i += 256) {
          const unsigned lo = 2u * i - (i & (stride - 1));
          const unsigned hi = lo + stride;
          const bool asc = ((lo & size) == 0u);
          const u64 a = keys[lo], b = keys[hi];
          if ((a > b) == asc) { keys[lo] = b; keys[hi] = a; }
        }
      }
    __syncthreads();

    // Serial union-find (elder rule identical to reference).
    if (t == 0) {
      for (int i = 0; i < EDGES; i++) {
        const u64 k = keys[i];
        const int u = (int)((k >> 8) & 255u), v = (int)(k & 255u);
        int ru = u;
        while (parent[ru] != ru) { const int pp = parent[parent[ru]]; parent[ru] = pp; ru = pp; }
        int rv = v;
        while (parent[rv] != rv) { const int pp = parent[parent[rv]]; parent[rv] = pp; rv = pp; }
        if (ru != rv) {
          const float fu = lg[ru], fv = lg[rv];
          const bool uel = (fu < fv) || ((fu == fv) && (ru < rv));
          const int elder = uel ? ru : rv;
          const int young = uel ? rv : ru;
          parent[young] = elder;
          death[young] = fmaxf(lg[u], lg[v]);
          paired[young] = 1;
        }
      }
    }
    __syncthreads();

    // Final root per node + per-component max (order-independent via uint max).
    int r = t;
    while (parent[r] != r) r = parent[r];
    atomicMax(&cmax[r], ordKey(lg[t]));
    __syncthreads();
    dpair[pass][t] = paired[t] ? death[t] : ordInv(cmax[r]);
    __syncthreads();
  }

  // phi MLPs + sum pooling: thread t owns hidden feature j = t.
  {
    const float a00 = w_phi0[t], a10 = w_phi0[NODES + t], bb0 = b_phi0[t];
    const float a01 = w_phi1[t], a11 = w_phi1[NODES + t], bb1 = b_phi1[t];
    float p0 = 0.f, p1 = 0.f;
    for (int n = 0; n < NODES; n++) {
      const float fn = lf[n];
      p0 += fmaxf(fn * a00 + dpair[0][n] * a10 + bb0, 0.f);          // h0=(f, d_sub)
      p1 += fmaxf((-dpair[1][n]) * a01 + fn * a11 + bb1, 0.f);       // h1=(-d_sup, f)
    }
    pbuf[0][t] = p0; pbuf[1][t] = p1;
  }
  __syncthreads();

  // Linear head, deterministic fixed-order dot per class.
  if (t < 10) {
    float acc = b_head[t];
    for (int j = 0; j < NODES; j++) {
      acc += pbuf[0][j] * w_head[j * 10 + t];
      acc += pbuf[1][j] * w_head[(NODES + j) * 10 + t];
    }
    out[g * 10 + t] = acc;
  }
}

// ---------------------------------------------------------------------------
extern "C" void kernel_launch(void* const* d_in, const int* in_sizes, int n_in,
                              void* d_out, int out_size, void* d_ws, size_t ws_size,
                              hipStream_t stream) {
  const float* x      = (const float*)d_in[0];
  const int*   edges  = (const int*)d_in[1];
  // d_in[2] sample_id unused: node->graph mapping is implicit (256 nodes/graph)
  const float* w_fil1 = (const float*)d_in[3];
  const float* b_fil1 = (const float*)d_in[4];
  const float* w_fil2 = (const float*)d_in[5];
  const float* b_fil2 = (const float*)d_in[6];
  const float* w_phi0 = (const float*)d_in[7];
  const float* b_phi0 = (const float*)d_in[8];
  const float* w_phi1 = (const float*)d_in[9];
  const float* b_phi1 = (const float*)d_in[10];
  const float* w_head = (const float*)d_in[11];
  const float* b_head = (const float*)d_in[12];

  float* f = (float*)d_ws;  // [16384] node filtration values

  const size_t smem_bytes = (size_t)(ROWS * APAD + WAVES * ROWS) * sizeof(float);
  fil_gemm_sigmoid<<<(NUM_GRAPHS * NODES) / ROWS, 256, smem_bytes, stream>>>(
      x, w_fil1, b_fil1, w_fil2, b_fil2, f);

  persistence_cls<<<NUM_GRAPHS, 256, 0, stream>>>(
      f, edges, w_phi0, b_phi0, w_phi1, b_phi1, w_head, b_head, (float*)d_out);
}